// TokenPairRepresentation_36807869727253
// MI455X (gfx1250) — compile-verified
//
#include <hip/hip_runtime.h>
#include <hip/hip_bf16.h>

// Shapes
#define BB   4
#define SS   32
#define NN   1024      // S*S
#define HIN  256
#define HID  128
#define HEADS 8
#define WCOLS 3072     // HID*OUT*3

typedef __attribute__((ext_vector_type(2))) float v2f;
typedef __attribute__((ext_vector_type(4))) float v4f;
typedef __attribute__((ext_vector_type(8))) float v8f;

__device__ __forceinline__ v8f wmma_f32(v2f a, v2f b, v8f c) {
  // V_WMMA_F32_16X16X4_F32 : (neg_a, A, neg_b, B, c_mod, C, reuse_a, reuse_b)
  return __builtin_amdgcn_wmma_f32_16x16x4_f32(false, a, false, b, (short)0, c,
                                               false, false);
}

__device__ __forceinline__ float wave_max(float v) {
#pragma unroll
  for (int off = 16; off > 0; off >>= 1) v = fmaxf(v, __shfl_xor(v, off, 32));
  return v;
}
__device__ __forceinline__ float wave_sum(float v) {
#pragma unroll
  for (int off = 16; off > 0; off >>= 1) v += __shfl_xor(v, off, 32);
  return v;
}

// ---------------------------------------------------------------------------
// Kernel 1: Y1[b] = x[b] @ W[0:256,:] + bias ; Y2[b] = x[b] @ W[256:512,:]
// One wave per 16x16 output tile, K=256 via f32 WMMA (16x16x4).
// waves = B(4) * ymat(2) * mtile(2) * ntile(192) = 3072
// ---------------------------------------------------------------------------
__global__ __launch_bounds__(128) void tpr_qkv_gemm(
    const float* __restrict__ x, const float* __restrict__ W,
    const float* __restrict__ bias, float* __restrict__ Y1,
    float* __restrict__ Y2) {
  int wid = (blockIdx.x * blockDim.x + threadIdx.x) >> 5;
  int lane = threadIdx.x & 31;
  int nt = wid % 192; wid /= 192;
  int mt = wid & 1;  wid >>= 1;
  int ym = wid & 1;  wid >>= 1;
  int bb = wid;                               // 0..3
  int khi = (lane >> 4) << 1;                 // 0 or 2
  const float* xrow  = x + ((size_t)(bb * SS + mt * 16 + (lane & 15))) * HIN;
  const float* Wbase = W + (ym ? (size_t)HIN * WCOLS : 0);
  int col = nt * 16 + (lane & 15);
  v8f c = {};
  for (int k0 = 0; k0 < HIN; k0 += 4) {
    v2f a = *(const v2f*)(xrow + k0 + khi);
    v2f bf;
    bf.x = Wbase[(size_t)(k0 + khi) * WCOLS + col];
    bf.y = Wbase[(size_t)(k0 + khi + 1) * WCOLS + col];
    c = wmma_f32(a, bf, c);
  }
  float* Y = ym ? Y2 : Y1;
  float bv = ym ? 0.0f : bias[col];
#pragma unroll
  for (int r = 0; r < 8; ++r) {
    int m = mt * 16 + r + ((lane >> 4) << 3);
    Y[((size_t)(bb * SS + m)) * WCOLS + col] = c[r] + bv;
  }
}

// ---------------------------------------------------------------------------
// Kernel 2: G matrices per (b,h): G11=q1·k1ᵀ G12=q1·k2ᵀ G21=q2·k1ᵀ G22=q2·k2ᵀ
// each 32x32, K = HID = 128, scaled by 1/sqrt(HID).
// waves = B*H*4g*4tiles = 512
// ---------------------------------------------------------------------------
__global__ __launch_bounds__(128) void tpr_gmat(
    const float* __restrict__ Y1, const float* __restrict__ Y2,
    float* __restrict__ G) {
  int wid = (blockIdx.x * blockDim.x + threadIdx.x) >> 5;
  int lane = threadIdx.x & 31;
  int nt = wid & 1; wid >>= 1;
  int mt = wid & 1; wid >>= 1;
  int g  = wid & 3; wid >>= 2;
  int h  = wid & 7; wid >>= 3;
  int bb = wid;
  const float* qm = (g < 2 ? Y1 : Y2) + (size_t)bb * SS * WCOLS + h * 384;
  const float* km = ((g & 1) == 0 ? Y1 : Y2) + (size_t)bb * SS * WCOLS + h * 384 + HID;
  int khi = (lane >> 4) << 1;
  const float* qrow = qm + (size_t)(mt * 16 + (lane & 15)) * WCOLS;
  const float* krow = km + (size_t)(nt * 16 + (lane & 15)) * WCOLS;
  v8f c = {};
  for (int k0 = 0; k0 < HID; k0 += 4) {
    v2f a  = *(const v2f*)(qrow + k0 + khi);
    v2f bf = *(const v2f*)(krow + k0 + khi);   // B[k,n] = k[n,k] (kᵀ)
    c = wmma_f32(a, bf, c);
  }
  const float scale = 0.088388347648318447f;   // 1/sqrt(128)
  float* Gout = G + ((size_t)((bb * HEADS + h) * 4 + g)) * 1024;
#pragma unroll
  for (int r = 0; r < 8; ++r) {
    int m = mt * 16 + r + ((lane >> 4) << 3);
    int n = nt * 16 + (lane & 15);
    Gout[m * 32 + n] = c[r] * scale;
  }
}

// ---------------------------------------------------------------------------
// Kernel 3: fused logits + mask + softmax + score writeout + marginals P1,P2.
// Block = 256 thr (8 waves) handles one (b,h,m1): 32 rows M=(m1,n1).
// Lane l owns m2=l, n2=0..31: logit = (Rm[l]+Rn[n2])*pm - (1-pm)*1e12.
// Score rows leave via non-temporal b128 stores (write-once 128MB stream).
// ---------------------------------------------------------------------------
#define LP 33
__global__ __launch_bounds__(256) void tpr_softmax(
    const float* __restrict__ G, const float* __restrict__ mask,
    float* __restrict__ score, float* __restrict__ P1,
    float* __restrict__ P2) {
  __shared__ float g11[32], g12[32], g21[1024], g22[1024], mk[32];
  __shared__ float tbuf[8][32 * LP];
  int bid = blockIdx.x;
  int m1 = bid & 31;
  int bh = bid >> 5;              // b*8+h
  int bb = bh >> 3;
  int tid = threadIdx.x, lane = tid & 31, wave = tid >> 5;
  const float* Gb = G + (size_t)bh * 4 * 1024;
  for (int i = tid; i < 1024; i += 256) {
    g21[i] = Gb[2 * 1024 + i];
    g22[i] = Gb[3 * 1024 + i];
  }
  if (tid < 32) {
    g11[tid] = Gb[m1 * 32 + tid];
    g12[tid] = Gb[1024 + m1 * 32 + tid];
    mk[tid]  = mask[bb * SS + tid];
  }
  __syncthreads();
  float mlane = mk[lane];
  float* tb = tbuf[wave];
#pragma unroll 1
  for (int r = 0; r < 4; ++r) {
    int n1 = wave + 8 * r;
    int M = m1 * 32 + n1;
    float Rm = g11[lane] + g21[n1 * 32 + lane];
    float vals[32];
    float vmax = -3.0e38f;
#pragma unroll
    for (int n2 = 0; n2 < 32; ++n2) {
      float Rn = g12[n2] + g22[n1 * 32 + n2];
      float pm = mlane * mk[n2];
      float v = (Rm + Rn) * pm - (1.0f - pm) * 1.0e12f;
      vals[n2] = v;
      vmax = fmaxf(vmax, v);
    }
    vmax = wave_max(vmax);
    float lsum = 0.0f;
#pragma unroll
    for (int n2 = 0; n2 < 32; ++n2) {
      float e = __expf(vals[n2] - vmax);
      vals[n2] = e;
      lsum += e;
    }
    float inv = 1.0f / wave_sum(lsum);
    float* srow = score + ((size_t)bh * NN + M) * NN + lane * 32;
    float p1 = 0.0f;
#pragma unroll
    for (int n2 = 0; n2 < 32; ++n2) {
      float s = vals[n2] * inv;
      vals[n2] = s;
      p1 += s;
      tb[lane * LP + n2] = s;
    }
#pragma unroll
    for (int n2 = 0; n2 < 32; n2 += 4) {   // 8 x b128 NT stores, 16B aligned
      v4f sv = {vals[n2], vals[n2 + 1], vals[n2 + 2], vals[n2 + 3]};
      __builtin_nontemporal_store(sv, (v4f*)(srow + n2));
    }
    P1[((size_t)bh * NN + M) * 32 + lane] = p1;
    __syncthreads();
    float p2 = 0.0f;
#pragma unroll
    for (int m2 = 0; m2 < 32; ++m2) p2 += tb[m2 * LP + lane];
    P2[((size_t)bh * NN + M) * 32 + lane] = p2;
    __syncthreads();
  }
}

// ---------------------------------------------------------------------------
// Kernel 4: context = P1 @ v1 + P2 @ v2  per (b,h): (N x 32)@(32 x 128).
// waves = B*H * Mtiles(64) * Dtiles(8) = 16384; context leaves via NT stores.
// ---------------------------------------------------------------------------
__global__ __launch_bounds__(256) void tpr_context(
    const float* __restrict__ P1, const float* __restrict__ P2,
    const float* __restrict__ Y1, const float* __restrict__ Y2,
    float* __restrict__ ctx) {
  int wid = (blockIdx.x * blockDim.x + threadIdx.x) >> 5;
  int lane = threadIdx.x & 31;
  int dt = wid & 7;  wid >>= 3;
  int mt = wid & 63; wid >>= 6;
  int h  = wid & 7;  wid >>= 3;
  int bb = wid;
  int bh = bb * HEADS + h;
  size_t pbase = ((size_t)bh * NN + mt * 16) * 32;
  const float* p1 = P1 + pbase;
  const float* p2 = P2 + pbase;
  const float* v1 = Y1 + (size_t)bb * SS * WCOLS + h * 384 + 2 * HID;
  const float* v2 = Y2 + (size_t)bb * SS * WCOLS + h * 384 + 2 * HID;
  int khi = (lane >> 4) << 1;
  int rowA = lane & 15;
  int col = dt * 16 + (lane & 15);
  v8f c = {};
  for (int k0 = 0; k0 < 32; k0 += 4) {
    v2f a = *(const v2f*)(p1 + rowA * 32 + k0 + khi);
    v2f bf;
    bf.x = v1[(size_t)(k0 + khi) * WCOLS + col];
    bf.y = v1[(size_t)(k0 + khi + 1) * WCOLS + col];
    c = wmma_f32(a, bf, c);
  }
  for (int k0 = 0; k0 < 32; k0 += 4) {
    v2f a = *(const v2f*)(p2 + rowA * 32 + k0 + khi);
    v2f bf;
    bf.x = v2[(size_t)(k0 + khi) * WCOLS + col];
    bf.y = v2[(size_t)(k0 + khi + 1) * WCOLS + col];
    c = wmma_f32(a, bf, c);
  }
#pragma unroll
  for (int r = 0; r < 8; ++r) {
    int m = mt * 16 + r + ((lane >> 4) << 3);
    __builtin_nontemporal_store(c[r], ctx + ((size_t)bh * NN + m) * HID + col);
  }
}

// ---------------------------------------------------------------------------
extern "C" void kernel_launch(void* const* d_in, const int* in_sizes, int n_in,
                              void* d_out, int out_size, void* d_ws,
                              size_t ws_size, hipStream_t stream) {
  const float* x    = (const float*)d_in[0];
  const float* mask = (const float*)d_in[1];
  const float* W    = (const float*)d_in[2];
  const float* bias = (const float*)d_in[3];

  float* ws = (float*)d_ws;
  float* Y1 = ws;                                 // B*32*3072 = 393216
  float* Y2 = Y1 + (size_t)BB * SS * WCOLS;       // 393216
  float* G  = Y2 + (size_t)BB * SS * WCOLS;       // B*H*4*1024 = 131072
  float* P1 = G  + (size_t)BB * HEADS * 4 * 1024; // B*H*N*32 = 1048576
  float* P2 = P1 + (size_t)BB * HEADS * NN * 32;  // 1048576

  float* ctx   = (float*)d_out;                   // B*H*N*128
  float* score = ctx + (size_t)BB * HEADS * NN * HID;

  // K1: 3072 waves
  tpr_qkv_gemm<<<768, 128, 0, stream>>>(x, W, bias, Y1, Y2);
  // K2: 512 waves
  tpr_gmat<<<128, 128, 0, stream>>>(Y1, Y2, G);
  // K3: one block per (b,h,m1)
  tpr_softmax<<<BB * HEADS * SS, 256, 0, stream>>>(G, mask, score, P1, P2);
  // K4: 16384 waves
  tpr_context<<<2048, 256, 0, stream>>>(P1, P2, Y1, Y2, ctx);
}